// AttnDecoder_14250701488235
// MI455X (gfx1250) — compile-verified
//
#include <hip/hip_runtime.h>
#include <math.h>

// ---------------------------------------------------------------------------
// AttnDecoder single step, MI455X (gfx1250, wave32).
// Memory-bound (~550MB f32 weight reads -> ~24us floor at 23.3 TB/s).
// All GEMVs run through V_WMMA_F32_16X16X4_F32 (exact f32 FMA semantics),
// one 16-row tile per wave, chained f32 accumulation in the 8-VGPR D matrix.
// ---------------------------------------------------------------------------

#define H  2048
#define V  50257
#define L  128

typedef __attribute__((ext_vector_type(2))) float v2f;
typedef __attribute__((ext_vector_type(8))) float v8f;

// One wave computes y[tile*16 .. tile*16+15] = act(W[rows,:] @ x + b).
// A-matrix layout (ISA 7.12.2, 32-bit A 16x4): lanes 0-15 -> M=lane,K={0,1};
// lanes 16-31 -> M=lane-16,K={2,3}. B replicated across columns so every
// column of D equals the GEMV result; lane 0 / lane 16 hold the 16 outputs.
__device__ __forceinline__ void gemv_tile16_wmma(
    const float* __restrict__ W, const float* __restrict__ x,
    const float* __restrict__ bias, float* __restrict__ y,
    int nRows, int K, int tile, int relu)
{
    const int lane = threadIdx.x & 31;
    const int half = lane >> 4;          // 0: K={0,1}, 1: K={2,3}
    int row = (tile << 4) + (lane & 15);
    int rc  = row < nRows ? row : (nRows - 1);   // clamp loads, keep EXEC full

    const float* wp = W + (size_t)rc * (size_t)K + 2 * half;
    const float* xp = x + 2 * half;

    v8f acc = {};
    #pragma unroll 4
    for (int k = 0; k < K; k += 4) {
        v2f a = *(const v2f*)(wp + k);   // weight row chunk (per-lane)
        v2f b = *(const v2f*)(xp + k);   // x chunk (broadcast within half)
        acc = __builtin_amdgcn_wmma_f32_16x16x4_f32(
            false, a, false, b, (short)0, acc, false, false);
    }

    // D layout: lane<16 vgpr j = D[M=j][N=lane]; lane>=16 vgpr j = D[M=8+j][N=lane-16]
    if ((lane & 15) == 0) {
        int base = (tile << 4) + (half << 3);
        #pragma unroll
        for (int j = 0; j < 8; ++j) {
            int r = base + j;
            if (r < nRows) {
                float v = acc[j] + bias[r];
                if (relu) v = fmaxf(v, 0.0f);
                y[r] = v;
            }
        }
    }
}

__global__ void gemv16_kernel(const float* __restrict__ W,
                              const float* __restrict__ x,
                              const float* __restrict__ bias,
                              float* __restrict__ y,
                              int nRows, int K, int relu)
{
    int wave = threadIdx.x >> 5;
    int tile = blockIdx.x * (blockDim.x >> 5) + wave;
    int nTiles = (nRows + 15) >> 4;
    if (tile >= nTiles) return;          // wave-uniform exit (EXEC all-1 below)
    gemv_tile16_wmma(W, x, bias, y, nRows, K, tile, relu);
}

// embedded = emb[idx]; build attn input [emb, h0] and first half of comb input
__global__ void prep_kernel(const int* __restrict__ idx,
                            const float* __restrict__ emb,
                            const float* __restrict__ hidden,
                            float* __restrict__ x_attn,
                            float* __restrict__ comb_in)
{
    int i = blockIdx.x * blockDim.x + threadIdx.x;
    if (i < H) {
        float e = emb[(size_t)idx[0] * H + i];
        x_attn[i]     = e;
        x_attn[H + i] = hidden[i];
        comb_in[i]    = e;
    }
}

__global__ void softmax128_kernel(const float* __restrict__ logits,
                                  float* __restrict__ w_out)
{
    __shared__ float sh[L];
    int t = threadIdx.x;
    float v = logits[t];
    sh[t] = v; __syncthreads();
    for (int s = 64; s > 0; s >>= 1) {
        if (t < s) sh[t] = fmaxf(sh[t], sh[t + s]);
        __syncthreads();
    }
    float m = sh[0]; __syncthreads();
    float e = expf(v - m);
    sh[t] = e; __syncthreads();
    for (int s = 64; s > 0; s >>= 1) {
        if (t < s) sh[t] += sh[t + s];
        __syncthreads();
    }
    w_out[t] = e / sh[0];
}

// attn_applied[h] = sum_l w[l] * enc[l,h]  -> comb_in[H + h]
__global__ void attn_apply_kernel(const float* __restrict__ w,
                                  const float* __restrict__ enc,
                                  float* __restrict__ comb_in)
{
    int h = blockIdx.x * blockDim.x + threadIdx.x;
    if (h < H) {
        float s = 0.0f;
        #pragma unroll 8
        for (int l = 0; l < L; ++l) s = fmaf(w[l], enc[(size_t)l * H + h], s);
        comb_in[H + h] = s;
    }
}

__device__ __forceinline__ float sigmoidf(float x) {
    return 1.0f / (1.0f + expf(-x));
}

// PyTorch GRU gate order r,z,n; biases already folded into gi/gh by gemv16.
__global__ void gru_combine_kernel(const float* __restrict__ gi,
                                   const float* __restrict__ gh,
                                   const float* __restrict__ h0,
                                   float* __restrict__ hnew)
{
    int h = blockIdx.x * blockDim.x + threadIdx.x;
    if (h < H) {
        float r = sigmoidf(gi[h]         + gh[h]);
        float z = sigmoidf(gi[H + h]     + gh[H + h]);
        float n = tanhf(  gi[2 * H + h] + r * gh[2 * H + h]);
        hnew[h] = (1.0f - z) * n + z * h0[h];
    }
}

// single-block max + log-sum-exp over V logits (200KB read, negligible)
__global__ void lse_kernel(const float* __restrict__ logits,
                           float* __restrict__ stats)
{
    __shared__ float sh[1024];
    int t = threadIdx.x;
    float m = -INFINITY;
    for (int i = t; i < V; i += 1024) m = fmaxf(m, logits[i]);
    sh[t] = m; __syncthreads();
    for (int s = 512; s > 0; s >>= 1) {
        if (t < s) sh[t] = fmaxf(sh[t], sh[t + s]);
        __syncthreads();
    }
    m = sh[0]; __syncthreads();
    float a = 0.0f;
    for (int i = t; i < V; i += 1024) a += expf(logits[i] - m);
    sh[t] = a; __syncthreads();
    for (int s = 512; s > 0; s >>= 1) {
        if (t < s) sh[t] += sh[t + s];
        __syncthreads();
    }
    if (t == 0) { stats[0] = m; stats[1] = logf(sh[0]); }
}

__global__ void logsoftmax_apply_kernel(const float* __restrict__ logits,
                                        const float* __restrict__ stats,
                                        float* __restrict__ out)
{
    int i = blockIdx.x * blockDim.x + threadIdx.x;
    if (i < V) out[i] = logits[i] - stats[0] - stats[1];
}

extern "C" void kernel_launch(void* const* d_in, const int* in_sizes, int n_in,
                              void* d_out, int out_size, void* d_ws, size_t ws_size,
                              hipStream_t stream)
{
    const int*   input_idx = (const int*)  d_in[0];
    const float* hidden    = (const float*)d_in[1];
    const float* enc       = (const float*)d_in[2];
    const float* emb       = (const float*)d_in[3];
    const float* attn_W    = (const float*)d_in[4];
    const float* attn_b    = (const float*)d_in[5];
    const float* comb_W    = (const float*)d_in[6];
    const float* comb_b    = (const float*)d_in[7];
    const float* W_ih      = (const float*)d_in[8];
    const float* W_hh      = (const float*)d_in[9];
    const float* b_ih      = (const float*)d_in[10];
    const float* b_hh      = (const float*)d_in[11];
    const float* out_W     = (const float*)d_in[12];
    const float* out_b     = (const float*)d_in[13];

    float* out = (float*)d_out;
    float* log_probs    = out;           // [V]
    float* h_new        = out + V;       // [H]
    float* attn_weights = out + V + H;   // [L]

    float* ws        = (float*)d_ws;
    float* x_attn    = ws;               // [2H] = [emb, h0]
    float* comb_in   = ws + 4096;        // [2H] = [emb, attn_applied]
    float* logits128 = ws + 8192;        // [L]
    float* x_comb    = ws + 8320;        // [H]
    float* gi        = ws + 10368;       // [3H]
    float* gh        = ws + 16512;       // [3H]
    float* logitsV   = ws + 22656;       // [V]
    float* stats     = ws + 72928;       // [2]

    // 1) gather embedding, assemble concat inputs
    prep_kernel<<<8, 256, 0, stream>>>(input_idx, emb, hidden, x_attn, comb_in);

    // 2) attn logits: [L=128 rows, K=2H]  (8 tiles -> 1 block)
    gemv16_kernel<<<1, 256, 0, stream>>>(attn_W, x_attn, attn_b, logits128,
                                         L, 2 * H, 0);

    // 3) softmax over 128 -> attn_weights (also an output)
    softmax128_kernel<<<1, 128, 0, stream>>>(logits128, attn_weights);

    // 4) attn_applied = w @ encoder_outputs -> comb_in[H..)
    attn_apply_kernel<<<8, 256, 0, stream>>>(attn_weights, enc, comb_in);

    // 5) x = relu(comb_W @ comb_in + comb_b): [H rows, K=2H] (128 tiles)
    gemv16_kernel<<<16, 256, 0, stream>>>(comb_W, comb_in, comb_b, x_comb,
                                          H, 2 * H, 1);

    // 6,7) GRU gate pre-activations: [3H rows, K=H] each (384 tiles)
    gemv16_kernel<<<48, 256, 0, stream>>>(W_ih, x_comb, b_ih, gi, 3 * H, H, 0);
    gemv16_kernel<<<48, 256, 0, stream>>>(W_hh, hidden, b_hh, gh, 3 * H, H, 0);

    // 8) gate math -> h_new (output slot)
    gru_combine_kernel<<<8, 256, 0, stream>>>(gi, gh, hidden, h_new);

    // 9) vocab projection: [V rows, K=H] — 412MB stream, the dominant cost
    {
        int nTiles = (V + 15) >> 4;              // 3142
        int blocks = (nTiles + 7) >> 3;          // 393 (8 waves/block)
        gemv16_kernel<<<blocks, 256, 0, stream>>>(out_W, h_new, out_b, logitsV,
                                                  V, H, 0);
    }

    // 10,11) log_softmax over V
    lse_kernel<<<1, 1024, 0, stream>>>(logitsV, stats);
    logsoftmax_apply_kernel<<<(V + 255) / 256, 256, 0, stream>>>(logitsV, stats,
                                                                 log_probs);
}